// ConvLayer_32272384262229
// MI455X (gfx1250) — compile-verified
//
#include <hip/hip_runtime.h>
#include <hip/hip_bf16.h>

typedef float v2f __attribute__((ext_vector_type(2)));
typedef float v8f __attribute__((ext_vector_type(8)));

__device__ __forceinline__ void fadd_agent(float* p, float v) {
    __hip_atomic_fetch_add(p, v, __ATOMIC_RELAXED, __HIP_MEMORY_SCOPE_AGENT);
}

// ---------------------------------------------------------------------------
// 1) BatchNorm partial sums: per-block sum & sumsq of relu(x) per feature.
//    Wave32 shfl_xor butterflies -> one LDS round -> 16-thread final sum.
// ---------------------------------------------------------------------------
__global__ void bn_partial_kernel(const float* __restrict__ x,
                                  float* __restrict__ partial, int n) {
    float v[16];
#pragma unroll
    for (int k = 0; k < 16; ++k) v[k] = 0.f;

    const int stride = gridDim.x * blockDim.x;
    for (int i = blockIdx.x * blockDim.x + threadIdx.x; i < n; i += stride) {
        const float4* p = (const float4*)(x + (size_t)i * 8);
        float4 v0 = p[0];
        float4 v1 = p[1];
        float h;
        h = fmaxf(v0.x, 0.f); v[0] += h; v[8]  += h * h;
        h = fmaxf(v0.y, 0.f); v[1] += h; v[9]  += h * h;
        h = fmaxf(v0.z, 0.f); v[2] += h; v[10] += h * h;
        h = fmaxf(v0.w, 0.f); v[3] += h; v[11] += h * h;
        h = fmaxf(v1.x, 0.f); v[4] += h; v[12] += h * h;
        h = fmaxf(v1.y, 0.f); v[5] += h; v[13] += h * h;
        h = fmaxf(v1.z, 0.f); v[6] += h; v[14] += h * h;
        h = fmaxf(v1.w, 0.f); v[7] += h; v[15] += h * h;
    }

    // intra-wave butterfly reduction (wave32)
#pragma unroll
    for (int k = 0; k < 16; ++k) {
        float t = v[k];
        t += __shfl_xor(t, 1);
        t += __shfl_xor(t, 2);
        t += __shfl_xor(t, 4);
        t += __shfl_xor(t, 8);
        t += __shfl_xor(t, 16);
        v[k] = t;
    }

    __shared__ float sh[8 * 16];          // 8 waves per 256-thread block
    const int wave = threadIdx.x >> 5;
    const int lane = threadIdx.x & 31;
    if (lane == 0) {
#pragma unroll
        for (int k = 0; k < 16; ++k) sh[wave * 16 + k] = v[k];
    }
    __syncthreads();
    if (threadIdx.x < 16) {
        float t = 0.f;
#pragma unroll
        for (int w = 0; w < 8; ++w) t += sh[w * 16 + threadIdx.x];
        partial[blockIdx.x * 16 + threadIdx.x] = t;
    }
}

// ---------------------------------------------------------------------------
// 2) BN finalize: scale = gamma*rsqrt(var+eps), shift = beta - mean*scale
// ---------------------------------------------------------------------------
__global__ void bn_finalize_kernel(const float* __restrict__ partial,
                                   const float* __restrict__ gamma,
                                   const float* __restrict__ beta,
                                   float* __restrict__ scsh, int nb, float invN) {
    int k = threadIdx.x;
    if (k < 8) {
        float s = 0.f, q = 0.f;
        for (int b = 0; b < nb; ++b) {
            s += partial[b * 16 + k];
            q += partial[b * 16 + 8 + k];
        }
        float mean = s * invN;
        float var  = q * invN - mean * mean;   // biased variance
        float sc   = gamma[k] * rsqrtf(var + 1e-5f);
        scsh[k]     = sc;
        scsh[8 + k] = beta[k] - mean * sc;
    }
}

// ---------------------------------------------------------------------------
// 3) Degree init (self loops -> start at 1)
// ---------------------------------------------------------------------------
__global__ void deg_init_kernel(unsigned* __restrict__ deg, int n2) {
    int i = blockIdx.x * blockDim.x + threadIdx.x;
    if (i < n2) deg[i] = 1u;
}

// 4) Degree count with integer atomics
__global__ void deg_count_kernel(const int* __restrict__ rows,
                                 const int* __restrict__ cols,
                                 unsigned* __restrict__ deg_c,
                                 unsigned* __restrict__ deg_r, int ne) {
    int e = blockIdx.x * blockDim.x + threadIdx.x;
    if (e >= ne) return;
    atomicAdd(&deg_r[rows[e]], 1u);
    atomicAdd(&deg_c[cols[e]], 1u);
}

// 5) In-place deg (u32) -> dis = rsqrt(deg) (f32); deg >= 1 always
__global__ void deg_to_dis_kernel(unsigned* __restrict__ deg, int n2) {
    int i = blockIdx.x * blockDim.x + threadIdx.x;
    if (i >= n2) return;
    float f = (float)deg[i];
    ((float*)deg)[i] = rsqrtf(f);
}

// ---------------------------------------------------------------------------
// 6) Fused ReLU+BN + dual 8x8 GEMM via V_WMMA_F32_16X16X4_F32.
//    One wave32 computes one 16-row tile: D(16x16) = A(16x8) * B(8x16),
//    K split into two chained 16x16x4 WMMAs; B cols 8..15 are zero.
//    A layout: lane<16 holds K=ks+{0,1}, lane>=16 holds K=ks+{2,3} (row = l).
//    B layout: VGPR0 = rows ks / ks+2, VGPR1 = rows ks+1 / ks+3 (col = l).
//    D layout: VGPR j = rows j (lanes 0-15) / j+8 (lanes 16-31), col = l.
//    Stores: lanes 0-7 / 16-23 write 8 consecutive floats per row -> coalesced.
// ---------------------------------------------------------------------------
__global__ void transform_wmma_kernel(const float* __restrict__ x,
                                      const float* __restrict__ W_in,
                                      const float* __restrict__ W_out,
                                      const float* __restrict__ scsh,
                                      float* __restrict__ xw_in,
                                      float* __restrict__ xw_out,
                                      int nTiles, int n) {
    const int lane = threadIdx.x & 31;
    const int wave = threadIdx.x >> 5;
    const int tile = blockIdx.x * (blockDim.x >> 5) + wave;
    if (tile >= nTiles) return;          // wave-uniform: EXEC stays all-ones

    const int half = lane >> 4;          // 0 or 1
    const int l    = lane & 15;          // 0..15
    int row = tile * 16 + l;
    int rowc = row < n ? row : (n - 1);  // clamp loads for a partial last tile

    const int kA0 = 2 * half;            // K base for slice 0
    const int kA1 = 4 + 2 * half;        // K base for slice 1

    const float* xr = x + (size_t)rowc * 8;
    v2f a0, a1;
    a0.x = fmaxf(xr[kA0],     0.f) * scsh[kA0]     + scsh[8 + kA0];
    a0.y = fmaxf(xr[kA0 + 1], 0.f) * scsh[kA0 + 1] + scsh[8 + kA0 + 1];
    a1.x = fmaxf(xr[kA1],     0.f) * scsh[kA1]     + scsh[8 + kA1];
    a1.y = fmaxf(xr[kA1 + 1], 0.f) * scsh[kA1 + 1] + scsh[8 + kA1 + 1];

    v2f bi0 = {0.f, 0.f}, bi1 = {0.f, 0.f};
    v2f bo0 = {0.f, 0.f}, bo1 = {0.f, 0.f};
    if (l < 8) {                          // cols 8..15 of B are zero padding
        bi0.x = W_in[kA0 * 8 + l];       bi0.y = W_in[(kA0 + 1) * 8 + l];
        bi1.x = W_in[kA1 * 8 + l];       bi1.y = W_in[(kA1 + 1) * 8 + l];
        bo0.x = W_out[kA0 * 8 + l];      bo0.y = W_out[(kA0 + 1) * 8 + l];
        bo1.x = W_out[kA1 * 8 + l];      bo1.y = W_out[(kA1 + 1) * 8 + l];
    }

    v8f ci = {}; v8f co = {};
    ci = __builtin_amdgcn_wmma_f32_16x16x4_f32(false, a0, false, bi0, (short)0, ci, false, false);
    ci = __builtin_amdgcn_wmma_f32_16x16x4_f32(false, a1, false, bi1, (short)0, ci, false, false);
    co = __builtin_amdgcn_wmma_f32_16x16x4_f32(false, a0, false, bo0, (short)0, co, false, false);
    co = __builtin_amdgcn_wmma_f32_16x16x4_f32(false, a1, false, bo1, (short)0, co, false, false);

    const bool fullTile = (tile * 16 + 16) <= n;   // wave-uniform
    if (l < 8) {
        const int mbase = tile * 16 + 8 * half;    // first row this lane group writes
        float* pi = xw_in  + (size_t)mbase * 8 + l;
        float* po = xw_out + (size_t)mbase * 8 + l;
        if (fullTile) {
#pragma unroll
            for (int j = 0; j < 8; ++j) {
                pi[(size_t)j * 8] = ci[j];
                po[(size_t)j * 8] = co[j];
            }
        } else {
#pragma unroll
            for (int j = 0; j < 8; ++j) {
                if (mbase + j < n) {
                    pi[(size_t)j * 8] = ci[j];
                    po[(size_t)j * 8] = co[j];
                }
            }
        }
    }
}

// ---------------------------------------------------------------------------
// 7) Output init: bias + self-loop contribution (norm = dis[i]^2)
// ---------------------------------------------------------------------------
__global__ void out_init_kernel(const float* __restrict__ xw_in,
                                const float* __restrict__ xw_out,
                                const float* __restrict__ dis_c,
                                const float* __restrict__ dis_r,
                                const float* __restrict__ b_in,
                                const float* __restrict__ b_out,
                                float* __restrict__ out, int n) {
    int i = blockIdx.x * blockDim.x + threadIdx.x;
    if (i >= n) return;
    float nc = dis_c[i] * dis_c[i];
    float nr = dis_r[i] * dis_r[i];
    const float* xi = xw_in  + (size_t)i * 8;
    const float* xo = xw_out + (size_t)i * 8;
    float* o = out + (size_t)i * 16;
#pragma unroll
    for (int k = 0; k < 8; ++k) o[k] = b_in[k] + xi[k] * nc;
#pragma unroll
    for (int k = 0; k < 8; ++k) o[8 + k] = b_out[k] + xo[k] * nr;
}

// ---------------------------------------------------------------------------
// 8) Edge scatter:
//    in-conv : out[col][0:8]  += xw_in[row]  * dis_c[row]*dis_c[col]
//    out-conv: out[row][8:16] += xw_out[col] * dis_r[row]*dis_r[col]
// ---------------------------------------------------------------------------
__global__ void edge_scatter_kernel(const int* __restrict__ rows,
                                    const int* __restrict__ cols,
                                    const float* __restrict__ xw_in,
                                    const float* __restrict__ xw_out,
                                    const float* __restrict__ dis_c,
                                    const float* __restrict__ dis_r,
                                    float* __restrict__ out, int ne) {
    int e = blockIdx.x * blockDim.x + threadIdx.x;
    if (e >= ne) return;
    int r = rows[e];
    int c = cols[e];
    float ni = dis_c[r] * dis_c[c];
    float no = dis_r[r] * dis_r[c];
    const float4* pi = (const float4*)(xw_in  + (size_t)r * 8);
    const float4* po = (const float4*)(xw_out + (size_t)c * 8);
    float4 xi0 = pi[0], xi1 = pi[1];
    float4 xo0 = po[0], xo1 = po[1];
    float* oc  = out + (size_t)c * 16;      // in_x half
    float* orr = out + (size_t)r * 16 + 8;  // out_x half
    fadd_agent(&oc[0], xi0.x * ni);
    fadd_agent(&oc[1], xi0.y * ni);
    fadd_agent(&oc[2], xi0.z * ni);
    fadd_agent(&oc[3], xi0.w * ni);
    fadd_agent(&oc[4], xi1.x * ni);
    fadd_agent(&oc[5], xi1.y * ni);
    fadd_agent(&oc[6], xi1.z * ni);
    fadd_agent(&oc[7], xi1.w * ni);
    fadd_agent(&orr[0], xo0.x * no);
    fadd_agent(&orr[1], xo0.y * no);
    fadd_agent(&orr[2], xo0.z * no);
    fadd_agent(&orr[3], xo0.w * no);
    fadd_agent(&orr[4], xo1.x * no);
    fadd_agent(&orr[5], xo1.y * no);
    fadd_agent(&orr[6], xo1.z * no);
    fadd_agent(&orr[7], xo1.w * no);
}

extern "C" void kernel_launch(void* const* d_in, const int* in_sizes, int n_in,
                              void* d_out, int out_size, void* d_ws, size_t ws_size,
                              hipStream_t stream) {
    const float* x     = (const float*)d_in[0];   // [N,8]
    const int*   ei    = (const int*)d_in[1];     // [2,E] flat: rows then cols
    const float* gamma = (const float*)d_in[2];
    const float* beta  = (const float*)d_in[3];
    const float* W_in  = (const float*)d_in[4];
    const float* b_in  = (const float*)d_in[5];
    const float* W_out = (const float*)d_in[6];
    const float* b_out = (const float*)d_in[7];
    float* out = (float*)d_out;                   // [N,16]

    const int n  = in_sizes[0] / 8;
    const int ne = in_sizes[1] / 2;
    const int NB = 1024;                          // BN reduction blocks

    // Workspace layout (floats): xw_in[N*8] | xw_out[N*8] | dis_c[N] | dis_r[N]
    //                            | partial[NB*16] | scale[8] shift[8]   (~36 MB)
    float* ws      = (float*)d_ws;
    float* xw_in   = ws;
    float* xw_out  = xw_in + (size_t)n * 8;
    float* dis_c   = xw_out + (size_t)n * 8;      // also deg_c (u32) before conversion
    float* dis_r   = dis_c + n;                   // also deg_r (u32)
    float* partial = dis_r + n;
    float* scsh    = partial + (size_t)NB * 16;

    const int* rows = ei;
    const int* cols = ei + ne;

    bn_partial_kernel<<<NB, 256, 0, stream>>>(x, partial, n);
    bn_finalize_kernel<<<1, 32, 0, stream>>>(partial, gamma, beta, scsh, NB, 1.0f / (float)n);

    deg_init_kernel<<<(2 * n + 255) / 256, 256, 0, stream>>>((unsigned*)dis_c, 2 * n);
    deg_count_kernel<<<(ne + 255) / 256, 256, 0, stream>>>(rows, cols,
                                                           (unsigned*)dis_c, (unsigned*)dis_r, ne);
    deg_to_dis_kernel<<<(2 * n + 255) / 256, 256, 0, stream>>>((unsigned*)dis_c, 2 * n);

    const int nTiles = (n + 15) / 16;             // 16 rows per wave32
    const int wavesPerBlock = 8;                  // 256 threads
    transform_wmma_kernel<<<(nTiles + wavesPerBlock - 1) / wavesPerBlock, 256, 0, stream>>>(
        x, W_in, W_out, scsh, xw_in, xw_out, nTiles, n);

    out_init_kernel<<<(n + 255) / 256, 256, 0, stream>>>(xw_in, xw_out, dis_c, dis_r,
                                                         b_in, b_out, out, n);
    edge_scatter_kernel<<<(ne + 255) / 256, 256, 0, stream>>>(rows, cols, xw_in, xw_out,
                                                              dis_c, dis_r, out, ne);
}